// SE3EncoderDecoderQM9_35648228557434
// MI455X (gfx1250) — compile-verified
//
#include <hip/hip_runtime.h>
#include <math.h>

// ---------------------------------------------------------------------------
// SE(3)-equivariant graph attention (QM9-style) for MI455X / gfx1250.
//
// Roofline: ~1.3 GFLOP total, ~20 MB of input -> ~1us of HBM time at 23.3TB/s
// => latency bound. Design: 1 workgroup (2 wave32s) per graph, everything in
// LDS. fp32 WMMA (v_wmma_f32_16x16x4_f32) for all dense matmuls (bit-faithful
// to the fp32 reference), async global->LDS (ASYNCcnt) staging for the
// adjacency block, weights pre-staged zero-padded in LDS so WMMA B-operands
// are branch-free ds loads with no global-load wait in front of each WMMA.
// ---------------------------------------------------------------------------

#define NB   2048
#define NN   64
#define TT   11
#define DD   8
#define HH   4
#define NOUT 19
#define RADIUS 10.0f

typedef float v2f __attribute__((ext_vector_type(2)));
typedef float v8f __attribute__((ext_vector_type(8)));

__device__ __forceinline__ v8f wmma4(v2f a, v2f b, v8f c) {
  // D = A(16x4) * B(4x16) + C, fp32 in/out.
  return __builtin_amdgcn_wmma_f32_16x16x4_f32(false, a, false, b, (short)0, c,
                                               false, false);
}

// LDS byte-offset of a shared-memory pointer (addrspace(3) is 32-bit).
__device__ __forceinline__ unsigned lds_off(const void* p) {
  return (unsigned)(unsigned long long)(const __attribute__((address_space(3))) char*)p;
}

// O[row,n] (n<8) = (accum? O : 0) + sum_k S[row,k] * W[k,n]
// S: LDS, 64 rows, element (row,k) at S[row*sRS + k*sKS]
// Wlds: LDS, 8 rows x 16 cols (cols 8..15 zero-padded)
// O: LDS, element (row,n) at O[row*oRS + n*oKS]
// Wave w handles 16-row tiles 2w,2w+1. N padded 8->16, K=8 (2 k-steps).
__device__ __forceinline__ void mm64x8_8x8(const float* S, int sRS, int sKS,
                                           const float* Wlds,
                                           float* O, int oRS, int oKS,
                                           int accum, unsigned tid) {
  const int wave = (int)(tid >> 5);
  const int lane = (int)(tid & 31);
  const int half = lane >> 4;   // 0: lanes 0-15, 1: lanes 16-31
  const int lr   = lane & 15;
  // B fragments (loop-invariant across row tiles): A/B lanes 16-31 carry K+2.
  v2f bf0, bf1;
  {
    const int kb = 0 + 2 * half;
    bf0.x = Wlds[kb * 16 + lr];
    bf0.y = Wlds[(kb + 1) * 16 + lr];
  }
  {
    const int kb = 4 + 2 * half;
    bf1.x = Wlds[kb * 16 + lr];
    bf1.y = Wlds[(kb + 1) * 16 + lr];
  }
  for (int t = 0; t < 2; ++t) {
    const int r0 = (wave * 2 + t) * 16;
    v8f acc = {};
    {
      const int kb = 0 + 2 * half;
      v2f a;
      a.x = S[(r0 + lr) * sRS + kb * sKS];
      a.y = S[(r0 + lr) * sRS + (kb + 1) * sKS];
      acc = wmma4(a, bf0, acc);
    }
    {
      const int kb = 4 + 2 * half;
      v2f a;
      a.x = S[(r0 + lr) * sRS + kb * sKS];
      a.y = S[(r0 + lr) * sRS + (kb + 1) * sKS];
      acc = wmma4(a, bf1, acc);
    }
    if (lr < 8) {
      for (int r = 0; r < 8; ++r) {
        const int row = r0 + r + 8 * half;   // D: vgpr r -> M=r (lo) / r+8 (hi)
        float* p = &O[row * oRS + lr * oKS];
        *p = accum ? (*p + acc[r]) : acc[r];
      }
    }
  }
}

// h = x(64x11, K-padded to 12) @ W_emb(12x16 LDS, zero-padded) + b_emb
__device__ __forceinline__ void mm_embed(const float* S /*[64][12] LDS*/,
                                         const float* Wlds /*[12][16] LDS*/,
                                         const float* __restrict__ bvec,
                                         float* O /*[64][8] LDS*/, unsigned tid) {
  const int wave = (int)(tid >> 5);
  const int lane = (int)(tid & 31);
  const int half = lane >> 4;
  const int lr   = lane & 15;
  v2f bf[3];
  for (int ks = 0; ks < 3; ++ks) {
    const int kb = ks * 4 + 2 * half;
    bf[ks].x = Wlds[kb * 16 + lr];
    bf[ks].y = Wlds[(kb + 1) * 16 + lr];
  }
  for (int t = 0; t < 2; ++t) {
    const int r0 = (wave * 2 + t) * 16;
    v8f acc = {};
    for (int ks = 0; ks < 3; ++ks) {
      const int kb = ks * 4 + 2 * half;
      v2f a;
      a.x = S[(r0 + lr) * 12 + kb];
      a.y = S[(r0 + lr) * 12 + kb + 1];
      acc = wmma4(a, bf[ks], acc);
    }
    if (lr < 8) {
      for (int r = 0; r < 8; ++r) {
        const int row = r0 + r + 8 * half;
        O[row * 8 + lr] = acc[r] + bvec[lr];
      }
    }
  }
}

__global__ __launch_bounds__(64)
void se3_graph_kernel(const float* __restrict__ x,
                      const float* __restrict__ coords,
                      const unsigned char* __restrict__ mask,
                      const unsigned char* __restrict__ adj,
                      const float* __restrict__ W_emb,
                      const float* __restrict__ b_emb,
                      const float* __restrict__ Wq,
                      const float* __restrict__ Wk,
                      const float* __restrict__ Wv0,
                      const float* __restrict__ Wg,
                      const float* __restrict__ Wo0,
                      const float* __restrict__ Wo1,
                      const float* __restrict__ w_dist,
                      const float* __restrict__ gammav,
                      const float* __restrict__ betav,
                      float* __restrict__ pooled_out) {
  __shared__ float s_x[NN * 12];            // x, K-padded
  __shared__ float s_wemb[12 * 16];         // W_emb zero-padded (rows 11, cols>=8)
  __shared__ float s_wmat[12 * 128];        // {q,k,v,g,o0,o1} x 2 layers, 8x16 padded
  __shared__ float s_h[NN * DD];            // type-0 feats
  __shared__ float s_f1[NN * DD * 3];       // type-1 feats
  __shared__ float s_hn[NN * DD];
  __shared__ float s_q[NN * DD], s_k[NN * DD], s_v[NN * DD], s_g[NN * DD];
  __shared__ float s_av[NN * DD];           // attn @ v   (all heads)
  __shared__ float s_o1[NN * DD * 3];       // out1       (all heads)
  __shared__ float s_c[NN * 3];
  __shared__ float s_dist[NN * NN];
  __shared__ float s_attn[NN * NN];         // per-head scratch (row-private)
  __shared__ uint4 s_adj_v[NN * NN / 16];   // adjacency bytes (16B aligned)
  __shared__ unsigned s_pmlo[NN], s_pmhi[NN];
  __shared__ float s_hasn[NN];
  __shared__ float s_maskf[NN];

  const int b = blockIdx.x;
  const unsigned tid = threadIdx.x;         // 0..63, thread == node row
  const int i = (int)tid;

  // ---- async-stage adjacency block (4KB) global -> LDS (ASYNCcnt path) ----
  {
    const unsigned lbase = lds_off(s_adj_v);
    const unsigned char* ag = adj + (size_t)b * NN * NN;
#pragma unroll
    for (int it = 0; it < 4; ++it) {
      const unsigned off = (unsigned)(it * 1024 + (int)tid * 16);
      const unsigned lds_a = lbase + off;
      const unsigned long long ga = (unsigned long long)(ag + off);
      asm volatile("global_load_async_to_lds_b128 %0, %1, off"
                   :: "v"(lds_a), "v"(ga) : "memory");
    }
  }

  // ---- stage inputs & weights while the DMA is in flight ----
  const float* xg = x + (size_t)b * NN * TT;
  for (int kk = 0; kk < 12; ++kk)
    s_x[i * 12 + kk] = (kk < TT) ? xg[i * TT + kk] : 0.f;
  const float* cg = coords + (size_t)b * NN * 3;
  s_c[i * 3 + 0] = cg[i * 3 + 0];
  s_c[i * 3 + 1] = cg[i * 3 + 1];
  s_c[i * 3 + 2] = cg[i * 3 + 2];
  s_maskf[i] = mask[(size_t)b * NN + i] ? 1.f : 0.f;
  for (int kk = 0; kk < 24; ++kk) s_f1[i * 24 + kk] = 0.f;

  {
    const float* mats[6] = {Wq, Wk, Wv0, Wg, Wo0, Wo1};
#pragma unroll
    for (int m = 0; m < 12; ++m) {
      const float* src = mats[m % 6] + (m / 6) * 64;
      for (int idx = (int)tid; idx < 128; idx += 64) {
        const int r = idx >> 4, c = idx & 15;
        s_wmat[m * 128 + idx] = (c < 8) ? src[r * 8 + c] : 0.f;
      }
    }
    for (int idx = (int)tid; idx < 12 * 16; idx += 64) {
      const int r = idx >> 4, c = idx & 15;
      s_wemb[idx] = (r < TT && c < 8) ? W_emb[r * 8 + c] : 0.f;
    }
  }

  asm volatile("s_wait_asynccnt 0" ::: "memory");
  __syncthreads();

  // ---- h = x @ W_emb + b_emb (WMMA) ----
  mm_embed(s_x, s_wemb, b_emb, s_h, tid);

  // ---- pairwise dist + pair mask (row i per thread) ----
  {
    const float cx = s_c[i * 3], cy = s_c[i * 3 + 1], cz = s_c[i * 3 + 2];
    const unsigned char* arow = (const unsigned char*)s_adj_v + i * NN;
    const float mi = s_maskf[i];
    unsigned lo = 0u, hi = 0u;
    for (int j = 0; j < NN; ++j) {
      const float dx = s_c[j * 3] - cx;
      const float dy = s_c[j * 3 + 1] - cy;
      const float dz = s_c[j * 3 + 2] - cz;
      const float d = sqrtf(dx * dx + dy * dy + dz * dz);
      s_dist[i * NN + j] = d;
      const bool pm = (mi > 0.f) && (s_maskf[j] > 0.f) && (arow[j] != 0) &&
                      (d <= RADIUS) && (j != i);
      if (pm) { if (j < 32) lo |= 1u << j; else hi |= 1u << (j - 32); }
    }
    s_pmlo[i] = lo; s_pmhi[i] = hi;
    s_hasn[i] = (lo | hi) ? 1.f : 0.f;
  }
  __syncthreads();

  const float inv_sqrt_dh = 0.7071067811865475f;   // 1/sqrt(DH=2)

  for (int l = 0; l < 2; ++l) {
    const float* wl = s_wmat + l * 6 * 128;
    // ---- hn = LayerNorm(h + |f1|) ----
    {
      float t[DD];
      float m = 0.f;
      for (int d = 0; d < DD; ++d) {
        const float fa = s_f1[i * 24 + d * 3 + 0];
        const float fb = s_f1[i * 24 + d * 3 + 1];
        const float fc = s_f1[i * 24 + d * 3 + 2];
        t[d] = s_h[i * 8 + d] + sqrtf(fa * fa + fb * fb + fc * fc + 1e-8f);
        m += t[d];
      }
      m *= 0.125f;
      float vv = 0.f;
      for (int d = 0; d < DD; ++d) { const float e = t[d] - m; vv += e * e; }
      vv *= 0.125f;
      const float r = rsqrtf(vv + 1e-5f);
      for (int d = 0; d < DD; ++d)
        s_hn[i * 8 + d] = (t[d] - m) * r * gammav[l * 8 + d] + betav[l * 8 + d];
    }
    __syncthreads();

    // ---- Q/K/V/G projections (WMMA) ----
    mm64x8_8x8(s_hn, 8, 1, wl + 0 * 128, s_q, 8, 1, 0, tid);
    mm64x8_8x8(s_hn, 8, 1, wl + 1 * 128, s_k, 8, 1, 0, tid);
    mm64x8_8x8(s_hn, 8, 1, wl + 2 * 128, s_v, 8, 1, 0, tid);
    mm64x8_8x8(s_hn, 8, 1, wl + 3 * 128, s_g, 8, 1, 0, tid);
    __syncthreads();

    // ---- per-head attention: softmax + neighbor sums (row-parallel VALU) ----
    for (int hh = 0; hh < HH; ++hh) {
      const int c0 = hh * 2, c1 = hh * 2 + 1;
      const float wd = w_dist[l * HH + hh];
      // masked softmax over j for row i
      {
        const float q0 = s_q[i * 8 + c0], q1 = s_q[i * 8 + c1];
        const unsigned lo = s_pmlo[i], hi = s_pmhi[i];
        float mx = -1e9f;
        for (int j = 0; j < NN; ++j) {
          const bool pm = (j < 32) ? ((lo >> j) & 1u) : ((hi >> (j - 32)) & 1u);
          float lg = -1e9f;
          if (pm)
            lg = (q0 * s_k[j * 8 + c0] + q1 * s_k[j * 8 + c1]) * inv_sqrt_dh +
                 wd * s_dist[i * NN + j];
          s_attn[i * NN + j] = lg;
          mx = fmaxf(mx, lg);
        }
        float sum = 0.f;
        for (int j = 0; j < NN; ++j) {
          const float e = __expf(s_attn[i * NN + j] - mx);
          s_attn[i * NN + j] = e;
          sum += e;
        }
        const float scl = s_hasn[i] / sum;   // rows w/o neighbors -> attn = 0
        for (int j = 0; j < NN; ++j) s_attn[i * NN + j] *= scl;
      }
      // out0 = attn @ v ; out1 via  sum_j w*c_j*g_j - c_i * sum_j w*g_j,
      // with w = attn / max(dist,1e-6)  (u folded in, no [N,N,3] tensor)
      {
        float a0 = 0.f, a1 = 0.f;
        float g0s = 0.f, g1s = 0.f;
        float t00 = 0.f, t01 = 0.f, t02 = 0.f, t10 = 0.f, t11 = 0.f, t12 = 0.f;
        const float cx = s_c[i * 3], cy = s_c[i * 3 + 1], cz = s_c[i * 3 + 2];
        for (int j = 0; j < NN; ++j) {
          const float a = s_attn[i * NN + j];
          a0 += a * s_v[j * 8 + c0];
          a1 += a * s_v[j * 8 + c1];
          const float w = a / fmaxf(s_dist[i * NN + j], 1e-6f);
          const float g0 = s_g[j * 8 + c0] * w;
          const float g1 = s_g[j * 8 + c1] * w;
          g0s += g0; g1s += g1;
          const float px = s_c[j * 3], py = s_c[j * 3 + 1], pz = s_c[j * 3 + 2];
          t00 += g0 * px; t01 += g0 * py; t02 += g0 * pz;
          t10 += g1 * px; t11 += g1 * py; t12 += g1 * pz;
        }
        s_av[i * 8 + c0] = a0;
        s_av[i * 8 + c1] = a1;
        s_o1[i * 24 + c0 * 3 + 0] = t00 - cx * g0s;
        s_o1[i * 24 + c0 * 3 + 1] = t01 - cy * g0s;
        s_o1[i * 24 + c0 * 3 + 2] = t02 - cz * g0s;
        s_o1[i * 24 + c1 * 3 + 0] = t10 - cx * g1s;
        s_o1[i * 24 + c1 * 3 + 1] = t11 - cy * g1s;
        s_o1[i * 24 + c1 * 3 + 2] = t12 - cz * g1s;
      }
    }
    __syncthreads();

    // ---- h += (attn@v) @ Wo0 ; f1 += out1 @ Wo1 per spatial c (WMMA) ----
    mm64x8_8x8(s_av, 8, 1, wl + 4 * 128, s_h, 8, 1, 1, tid);
    for (int c = 0; c < 3; ++c)
      mm64x8_8x8(s_o1 + c, 24, 3, wl + 5 * 128, s_f1 + c, 24, 3, 1, tid);
    __syncthreads();
  }

  // ---- invariant readout + masked mean pool ----
  {
    const float mi = s_maskf[i];
    for (int d = 0; d < DD; ++d) {
      const float fa = s_f1[i * 24 + d * 3 + 0];
      const float fb = s_f1[i * 24 + d * 3 + 1];
      const float fc = s_f1[i * 24 + d * 3 + 2];
      s_hn[i * 8 + d] =
          (s_h[i * 8 + d] + sqrtf(fa * fa + fb * fb + fc * fc + 1e-8f)) * mi;
    }
  }
  __syncthreads();
  if (tid < 8) {
    float sum = 0.f, cnt = 0.f;
    for (int j = 0; j < NN; ++j) { sum += s_hn[j * 8 + tid]; cnt += s_maskf[j]; }
    pooled_out[(size_t)b * 8 + tid] = sum / fmaxf(cnt, 1.f);
  }
}

// out[2048x19] = pooled[2048x8] @ W_out[8x19] + b_out   (fp32 WMMA GEMM)
__global__ __launch_bounds__(32)
void readout_kernel(const float* __restrict__ pooled,
                    const float* __restrict__ W_out,
                    const float* __restrict__ b_out,
                    float* __restrict__ out) {
  const int lane = (int)(threadIdx.x & 31);
  const int half = lane >> 4;
  const int lr   = lane & 15;
  const int r0 = blockIdx.x * 16;
  // B fragments hoisted (uniform per block).
  v2f b00, b01, b10, b11;
  {
    const int kb = 0 + 2 * half;
    b00.x = W_out[kb * NOUT + lr];
    b00.y = W_out[(kb + 1) * NOUT + lr];
    const int n1 = 16 + lr;
    b10.x = (n1 < NOUT) ? W_out[kb * NOUT + n1] : 0.f;
    b10.y = (n1 < NOUT) ? W_out[(kb + 1) * NOUT + n1] : 0.f;
  }
  {
    const int kb = 4 + 2 * half;
    b01.x = W_out[kb * NOUT + lr];
    b01.y = W_out[(kb + 1) * NOUT + lr];
    const int n1 = 16 + lr;
    b11.x = (n1 < NOUT) ? W_out[kb * NOUT + n1] : 0.f;
    b11.y = (n1 < NOUT) ? W_out[(kb + 1) * NOUT + n1] : 0.f;
  }
  v8f acc0 = {}, acc1 = {};
  {
    const int kb = 0 + 2 * half;
    v2f a;
    a.x = pooled[(r0 + lr) * 8 + kb];
    a.y = pooled[(r0 + lr) * 8 + kb + 1];
    acc0 = wmma4(a, b00, acc0);
    acc1 = wmma4(a, b10, acc1);
  }
  {
    const int kb = 4 + 2 * half;
    v2f a;
    a.x = pooled[(r0 + lr) * 8 + kb];
    a.y = pooled[(r0 + lr) * 8 + kb + 1];
    acc0 = wmma4(a, b01, acc0);
    acc1 = wmma4(a, b11, acc1);
  }
  for (int r = 0; r < 8; ++r) {
    const int row = r0 + r + 8 * half;
    out[row * NOUT + lr] = acc0[r] + b_out[lr];
    const int n1 = 16 + lr;
    if (n1 < NOUT) out[row * NOUT + n1] = acc1[r] + b_out[n1];
  }
}

extern "C" void kernel_launch(void* const* d_in, const int* in_sizes, int n_in,
                              void* d_out, int out_size, void* d_ws,
                              size_t ws_size, hipStream_t stream) {
  (void)in_sizes; (void)n_in; (void)out_size; (void)ws_size;
  const float*         x      = (const float*)d_in[0];
  const float*         coords = (const float*)d_in[1];
  const unsigned char* mask   = (const unsigned char*)d_in[2];
  const unsigned char* adj    = (const unsigned char*)d_in[3];
  const float*         W_emb  = (const float*)d_in[4];
  const float*         b_emb  = (const float*)d_in[5];
  const float*         Wq     = (const float*)d_in[6];
  const float*         Wk     = (const float*)d_in[7];
  const float*         Wv0    = (const float*)d_in[8];
  const float*         Wg     = (const float*)d_in[9];
  const float*         Wo0    = (const float*)d_in[10];
  const float*         Wo1    = (const float*)d_in[11];
  const float*         w_dist = (const float*)d_in[12];
  const float*         gammav = (const float*)d_in[13];
  const float*         betav  = (const float*)d_in[14];
  const float*         W_out  = (const float*)d_in[15];
  const float*         b_out  = (const float*)d_in[16];

  float* pooled = (float*)d_ws;           // [2048 x 8] fp32 = 64 KB scratch
  float* out    = (float*)d_out;          // [2048 x 19] fp32

  se3_graph_kernel<<<dim3(NB), dim3(64), 0, stream>>>(
      x, coords, mask, adj, W_emb, b_emb, Wq, Wk, Wv0, Wg, Wo0, Wo1, w_dist,
      gammav, betav, pooled);
  readout_kernel<<<dim3(NB / 16), dim3(32), 0, stream>>>(pooled, W_out, b_out,
                                                         out);
}